// Augment_77532749627421
// MI455X (gfx1250) — compile-verified
//
#include <hip/hip_runtime.h>
#include <cstdint>

// out[b,h,w,d,f] = valid(h',w') ? x[b, iy, ix, 127-d, f] : 0
//   h' = (h-7)&127 ; w' = (140-w)&127 ; (iy,ix) = round_ne(rot80(h',w') about 63.5)
// One block per (b,h,w) output row: 1024 floats = 256 float4 = 4KB contiguous.
// Pure bandwidth kernel: ~256 MiB total traffic -> ~11 us at 23.3 TB/s.

typedef int v4i __attribute__((ext_vector_type(4)));

#if defined(__gfx1250__) && __has_builtin(__builtin_amdgcn_global_load_async_to_lds_b128) && __has_builtin(__builtin_amdgcn_global_store_async_from_lds_b128)
#define AUG_ASYNC_LDS 1
typedef __attribute__((address_space(3))) v4i lds_v4i;
#else
#define AUG_ASYNC_LDS 0
#endif

__device__ __forceinline__ void aug_wait_asynccnt0() {
#if defined(__gfx1250__)
#if __has_builtin(__builtin_amdgcn_s_wait_asynccnt)
    __builtin_amdgcn_s_wait_asynccnt(0);
#else
    asm volatile("s_wait_asynccnt 0x0" ::: "memory");
#endif
#endif
}

__global__ __launch_bounds__(256) void augment_rot_roll_flip(
        const float* __restrict__ x, float* __restrict__ out) {
    // Grid: 2*128*128 blocks, blockIdx.x = ((b*128)+h)*128 + w
    const int bhw = blockIdx.x;
    const int w = bhw & 127;
    const int h = (bhw >> 7) & 127;

    // Fold roll(7,-13) and flip(W) into the source pixel of the rotation.
    const int hp = (h - 7) & 127;     // (h-7) mod 128
    const int wp = (140 - w) & 127;   // (127-w+13) mod 128

    // Nearest-neighbor inverse-rotation sample about center 63.5 (round-half-even).
    const float c80 = 0.17364817766693041f;  // cos(80 deg)
    const float s80 = 0.98480775301220810f;  // sin(80 deg)
    const float dy = (float)hp - 63.5f;
    const float dx = (float)wp - 63.5f;
    const float sxf = c80 * dx - s80 * dy + 63.5f;
    const float syf = s80 * dx + c80 * dy + 63.5f;
    const int ix = __float2int_rn(sxf);
    const int iy = __float2int_rn(syf);
    const bool valid = ((unsigned)iy < 128u) && ((unsigned)ix < 128u);

    const int t = threadIdx.x;  // 0..255 : one float4 (16B) per lane
    float4* dst = reinterpret_cast<float4*>(out) + (size_t)bhw * 256 + t;

    if (!valid) {
        // Rotated outside the image: zero-fill the whole 4KB row.
        *dst = float4{0.f, 0.f, 0.f, 0.f};
        return;
    }

    // flip(D): out float4 t covers (d = t>>1, f-half = t&1); source float4 is
    // (127-d)*2 + (t&1). 32B granules stay contiguous; only d order reverses.
    const int b = bhw >> 14;
    const int src4 = ((127 - (t >> 1)) << 1) | (t & 1);
    const float4* src = reinterpret_cast<const float4*>(x)
        + (size_t)((b << 14) | (iy << 7) | ix) * 256 + src4;

#if AUG_ASYNC_LDS
    // CDNA5 async data mover: global -> LDS (d-reversal folded into per-lane
    // source address, LDS laid out in output order), then LDS -> global.
    // Both ops are issued by the same wave on its own LDS range, so a single
    // ASYNCcnt wait orders them; no workgroup barrier needed. No VGPR staging.
    __shared__ float4 lds[256];
    lds_v4i* lptr = (lds_v4i*)&lds[t];
    __builtin_amdgcn_global_load_async_to_lds_b128((v4i*)src, lptr, 0, 0);
    aug_wait_asynccnt0();
    __builtin_amdgcn_global_store_async_from_lds_b128((v4i*)dst, lptr, 0, 0);
    aug_wait_asynccnt0();
#else
    // Fallback: direct b128 copy through VGPRs.
    *dst = *src;
#endif
}

extern "C" void kernel_launch(void* const* d_in, const int* in_sizes, int n_in,
                              void* d_out, int out_size, void* d_ws, size_t ws_size,
                              hipStream_t stream) {
    const float* x = (const float*)d_in[0];
    float* out = (float*)d_out;
    // 2 * 128 * 128 = 32768 output (b,h,w) rows; 256 threads move 4KB each.
    dim3 grid(2 * 128 * 128);
    dim3 block(256);
    hipLaunchKernelGGL(augment_rot_roll_flip, grid, block, 0, stream, x, out);
}